// ConvBundle_48146583388363
// MI455X (gfx1250) — compile-verified
//
#include <hip/hip_runtime.h>

typedef __attribute__((ext_vector_type(2))) float v2f;
typedef __attribute__((ext_vector_type(8))) float v8f;

#define Bn   16
#define Hn   96
#define Wn   96
#define CIN  64
#define COUT 128
#define NPIX (Bn * Hn * Wn)                // 147456 pixels
#define NTAP 9
#define TILES_PER_IMG (Hn * Wn / 16)       // 576
#define TILES_PER_TAP (Bn * TILES_PER_IMG) // 9216
#define NWAVES (NTAP * TILES_PER_TAP)      // 82944

// ---------------- Stage 1a: s[b,i,j] = sum_c inputs[b,i,j,c] ----------------
__global__ __launch_bounds__(256) void chansum_kernel(const float* __restrict__ in,
                                                      float* __restrict__ s) {
    int p = blockIdx.x * blockDim.x + threadIdx.x;
    if (p >= NPIX) return;
    const float4* base = ((const float4*)in) + (size_t)p * (CIN / 4);
    float acc = 0.f;
#pragma unroll
    for (int k = 0; k < CIN / 4; ++k) {
        float4 v = base[k];
        acc += v.x + v.y + v.z + v.w;
    }
    s[p] = acc;
}

// ---------------- Stage 1b: wsum[t,o] = sum_c W[t,c,o] ----------------------
__global__ __launch_bounds__(256) void wsum_kernel(const float* __restrict__ W,
                                                   float* __restrict__ wsum) {
    int id = blockIdx.x * blockDim.x + threadIdx.x;  // t*COUT + o
    if (id >= NTAP * COUT) return;
    int t = id / COUT, o = id % COUT;
    const float* base = W + (size_t)t * COUT * COUT + o;
    float acc = 0.f;
#pragma unroll 8
    for (int c = 0; c < COUT; ++c) acc += base[(size_t)c * COUT];
    wsum[id] = acc;
}

// ---------------- Stage 2: rank-1 outer product via V_WMMA_F32_16X16X4_F32 --
// One wave per (tap, 16-pixel tile). A[16x4]: col K=0 holds the 16 shifted s
// values; B[4x16]: row K=0 holds a 16-wide wsum slice. D = A x B, exact f32.
// All loads are unconditional (clamped / lane&15 addressing) with arithmetic
// zero-selection, so the wave stays convergent: one load clause up front,
// then straight-line 8x (WMMA + 8 NT stores).
__global__ __launch_bounds__(256) void outer_kernel(const float* __restrict__ s,
                                                    const float* __restrict__ wsum,
                                                    float* __restrict__ out) {
    const int lane      = threadIdx.x & 31;
    const int waveInBlk = threadIdx.x >> 5;
    const int waveId    = blockIdx.x * 8 + waveInBlk;  // exact grid: < NWAVES

    const int t  = waveId / TILES_PER_TAP;
    int rem      = waveId % TILES_PER_TAP;
    const int b  = rem / TILES_PER_IMG;
    rem          = rem % TILES_PER_IMG;
    const int i  = rem / (Wn / 16);
    const int j0 = (rem % (Wn / 16)) * 16;

    const int dx = t / 3 - 1;
    const int dy = t % 3 - 1;

    const int lo  = lane & 15;     // column / M index this lane services
    const bool lowHalf = (lane < 16);

    // ---- A value: shifted, zero-padded s. Clamp coords, always load, then
    // select 0 if out of range or upper half-wave.
    const int ii = i + dx;
    const int jj = j0 + lo + dy;
    const bool inb = lowHalf && (ii >= 0) && (ii < Hn) && (jj >= 0) && (jj < Wn);
    const int iic = ii < 0 ? 0 : (ii > Hn - 1 ? Hn - 1 : ii);
    const int jjc = jj < 0 ? 0 : (jj > Wn - 1 ? Wn - 1 : jj);
    const float sload = s[((size_t)b * Hn + iic) * Wn + jjc];
    const float aval  = inb ? sload : 0.f;

    v2f A;
    A.x = aval;
    A.y = 0.f;

    // ---- Preload all 8 wsum slices (one clause of loads, single wait).
    const float* wrow = wsum + t * COUT;
    float bv[8];
#pragma unroll
    for (int ct = 0; ct < 8; ++ct) {
        const float w = wrow[ct * 16 + lo];   // all 32 lanes load (dup upper)
        bv[ct] = lowHalf ? w : 0.f;
    }

    // ---- Output tile base: element [t, b, i, j0 + M, o]
    const size_t obase = ((((size_t)t * Bn + b) * Hn + i) * Wn + j0) * COUT;
    const int M0 = (lane >> 4) * 8;  // D rows this lane holds: M0 + r
    float* const outLane = out + obase + (size_t)M0 * COUT + (size_t)lo;

#pragma unroll
    for (int ct = 0; ct < 8; ++ct) {
        v2f Bv;
        Bv.x = bv[ct];
        Bv.y = 0.f;

        v8f C = {};
        C = __builtin_amdgcn_wmma_f32_16x16x4_f32(
            /*neg_a=*/false, A, /*neg_b=*/false, Bv,
            /*c_mod=*/(short)0, C, /*reuse_a=*/false, /*reuse_b=*/false);

        float* const colp = outLane + ct * 16;
#pragma unroll
        for (int r = 0; r < 8; ++r) {
            // Streaming store: keep the 680MB write stream from evicting
            // the L2-resident s/wsum working set.
            __builtin_nontemporal_store(C[r], colp + (size_t)r * COUT);
        }
    }
}

extern "C" void kernel_launch(void* const* d_in, const int* in_sizes, int n_in,
                              void* d_out, int out_size, void* d_ws, size_t ws_size,
                              hipStream_t stream) {
    const float* in = (const float*)d_in[0];   // [16,96,96,64]
    const float* W  = (const float*)d_in[1];   // [3,3,128,128]
    float* out      = (float*)d_out;           // [3,3,16,96,96,128]

    float* s    = (float*)d_ws;                // NPIX floats
    float* wsum = s + NPIX;                    // NTAP*COUT floats

    chansum_kernel<<<(NPIX + 255) / 256, 256, 0, stream>>>(in, s);
    wsum_kernel<<<(NTAP * COUT + 255) / 256, 256, 0, stream>>>(W, wsum);

    // 82944 waves total, 8 waves (256 threads) per block -> 10368 blocks exact.
    outer_kernel<<<NWAVES / 8, 256, 0, stream>>>(s, wsum, out);
}